// DistillationLoss_13950053777685
// MI455X (gfx1250) — compile-verified
//
#include <hip/hip_runtime.h>
#include <math.h>

typedef __attribute__((ext_vector_type(2))) float v2f;
typedef __attribute__((ext_vector_type(8))) float v8f;

#define B_ROWS 1024
#define C_COLS 21841
#define EPSF   1e-8f
#define TAU    4.0f

// ---------------- block reduction helpers (blockDim.x == 256) ----------------
__device__ __forceinline__ float block_sum(float v, float* sh) {
  const int tid = threadIdx.x;
  sh[tid] = v; __syncthreads();
  #pragma unroll
  for (int s = 128; s > 0; s >>= 1) {
    if (tid < s) sh[tid] += sh[tid + s];
    __syncthreads();
  }
  float r = sh[0]; __syncthreads();
  return r;
}

__device__ __forceinline__ float block_max(float v, float* sh) {
  const int tid = threadIdx.x;
  sh[tid] = v; __syncthreads();
  #pragma unroll
  for (int s = 128; s > 0; s >>= 1) {
    if (tid < s) sh[tid] = fmaxf(sh[tid], sh[tid + s]);
    __syncthreads();
  }
  float r = sh[0]; __syncthreads();
  return r;
}

// ---------------- Stage A: per-row mean and centered norm for s, t, p ----------------
__global__ __launch_bounds__(256) void row_stats_kernel(
    const float* __restrict__ s, const float* __restrict__ t, const float* __restrict__ p,
    float* __restrict__ mean, float* __restrict__ norm) {
  __shared__ float sh[256];
  const int row = blockIdx.x;
  const int mat = blockIdx.y;
  const float* X = (mat == 0) ? s : ((mat == 1) ? t : p);
  const float* R = X + (size_t)row * C_COLS;
  float sum = 0.f, sq = 0.f;
  for (int c = threadIdx.x; c < C_COLS; c += 256) {
    float x = R[c];
    sum += x;
    sq  += x * x;
  }
  sum = block_sum(sum, sh);
  sq  = block_sum(sq, sh);
  if (threadIdx.x == 0) {
    float m = sum / (float)C_COLS;
    mean[mat * B_ROWS + row] = m;
    float cs = sq - (float)C_COLS * m * m;       // sum of centered squares
    norm[mat * B_ROWS + row] = sqrtf(fmaxf(cs, 0.f));
  }
}

// ---------------- Stage B: D = A * B^T via V_WMMA_F32_16X16X4_F32 ----------------
// Block tile 64(M) x 128(N); 8 waves, each owns a 32x32 output (2x2 WMMA tiles).
// Double-buffered LDS staging: global->regs for chunk k+1 issued before the WMMA
// burst on chunk k; regs drained to the other LDS buffer; ONE barrier/iteration.
#define TM  64
#define TN  128
#define KCH 32
#define LDT 36   // padded LDS row stride (floats) to dodge bank conflicts

__global__ __launch_bounds__(256) void gemm_nt_wmma(
    const float* __restrict__ A, const float* __restrict__ Bm,
    float* __restrict__ D, int K) {
  __shared__ __align__(16) float As[2][TM][LDT];
  __shared__ __align__(16) float Bs[2][TN][LDT];

  const int tid  = threadIdx.x;
  const int wid  = tid >> 5;
  const int lane = tid & 31;
  const int l16  = lane & 15;
  const int kh   = lane >> 4;            // 0 or 1 (K half per ISA 16x4 fp32 layout)
  const int rowBlk = blockIdx.y * TM;
  const int colBlk = blockIdx.x * TN;
  const int wRow = (wid >> 2) * 32;      // 2 waves along M
  const int wCol = (wid & 3) * 32;       // 4 waves along N

  v8f c00 = {}; v8f c01 = {}; v8f c10 = {}; v8f c11 = {};

  const int lr = tid >> 3;               // 0..31 : row within stage chunk
  const int lc = (tid & 7) * 4;          // 0..28 : 4 consecutive K per thread

  // ---- prologue: stage chunk 0 into buffer 0 ----
  {
    const bool full = KCH <= K;
    #pragma unroll
    for (int i = 0; i < 2; ++i) {
      const float* src = A + (size_t)(rowBlk + lr + 32 * i) * K + lc;
      #pragma unroll
      for (int q = 0; q < 4; ++q)
        As[0][lr + 32 * i][lc + q] = (full || lc + q < K) ? src[q] : 0.f;
    }
    #pragma unroll
    for (int i = 0; i < 4; ++i) {
      const float* src = Bm + (size_t)(colBlk + lr + 32 * i) * K + lc;
      #pragma unroll
      for (int q = 0; q < 4; ++q)
        Bs[0][lr + 32 * i][lc + q] = (full || lc + q < K) ? src[q] : 0.f;
    }
  }
  __syncthreads();

  int buf = 0;
  for (int k0 = 0; k0 < K; k0 += KCH) {
    const int  kn      = k0 + KCH;
    const bool hasNext = kn < K;

    // ---- issue global loads for chunk k+1 into registers (overlaps WMMA) ----
    float ra[2][4];
    float rb[4][4];
    if (hasNext) {
      const bool full = (kn + KCH) <= K;
      #pragma unroll
      for (int i = 0; i < 2; ++i) {
        const float* src = A + (size_t)(rowBlk + lr + 32 * i) * K + kn + lc;
        __builtin_prefetch(src + KCH, 0, 3);     // chunk after next -> global_prefetch_b8
        #pragma unroll
        for (int q = 0; q < 4; ++q)
          ra[i][q] = (full || kn + lc + q < K) ? src[q] : 0.f;
      }
      #pragma unroll
      for (int i = 0; i < 4; ++i) {
        const float* src = Bm + (size_t)(colBlk + lr + 32 * i) * K + kn + lc;
        __builtin_prefetch(src + KCH, 0, 3);
        #pragma unroll
        for (int q = 0; q < 4; ++q)
          rb[i][q] = (full || kn + lc + q < K) ? src[q] : 0.f;
      }
    }

    // ---- WMMA burst on current chunk ----
    #pragma unroll
    for (int kk = 0; kk < KCH; kk += 4) {
      const int ks = kk + 2 * kh;        // lanes 0-15: K=kk,kk+1 ; lanes 16-31: K=kk+2,kk+3
      v2f a0 = *(const v2f*)&As[buf][wRow      + l16][ks];
      v2f a1 = *(const v2f*)&As[buf][wRow + 16 + l16][ks];
      v2f b0 = *(const v2f*)&Bs[buf][wCol      + l16][ks];
      v2f b1 = *(const v2f*)&Bs[buf][wCol + 16 + l16][ks];
      c00 = __builtin_amdgcn_wmma_f32_16x16x4_f32(false, a0, false, b0, (short)0, c00, false, false);
      c01 = __builtin_amdgcn_wmma_f32_16x16x4_f32(false, a0, false, b1, (short)0, c01, false, false);
      c10 = __builtin_amdgcn_wmma_f32_16x16x4_f32(false, a1, false, b0, (short)0, c10, false, false);
      c11 = __builtin_amdgcn_wmma_f32_16x16x4_f32(false, a1, false, b1, (short)0, c11, false, false);
    }

    // ---- drain regs into the other buffer; single barrier per iteration ----
    if (hasNext) {
      const int nb = buf ^ 1;
      #pragma unroll
      for (int i = 0; i < 2; ++i)
        #pragma unroll
        for (int q = 0; q < 4; ++q)
          As[nb][lr + 32 * i][lc + q] = ra[i][q];
      #pragma unroll
      for (int i = 0; i < 4; ++i)
        #pragma unroll
        for (int q = 0; q < 4; ++q)
          Bs[nb][lr + 32 * i][lc + q] = rb[i][q];
    }
    __syncthreads();
    buf ^= 1;
  }

  // C/D layout: VGPR r, lanes 0-15 -> M=r, N=lane ; lanes 16-31 -> M=r+8, N=lane-16
  const int m0 = rowBlk + wRow + 8 * kh;
  const int n0 = colBlk + wCol + l16;
  #pragma unroll
  for (int r = 0; r < 8; ++r) {
    D[(size_t)(m0 + r)      * B_ROWS + n0]      = c00[r];
    D[(size_t)(m0 + r)      * B_ROWS + n0 + 16] = c01[r];
    D[(size_t)(m0 + r + 16) * B_ROWS + n0]      = c10[r];
    D[(size_t)(m0 + r + 16) * B_ROWS + n0 + 16] = c11[r];
  }
}

// ---------------- Stage C: L_c and L_p from the dot-product matrices ----------------
__global__ __launch_bounds__(256) void contrast_kernel(
    const float* __restrict__ D1, const float* __restrict__ D2,
    const float* __restrict__ mean, const float* __restrict__ norm,
    const int* __restrict__ lab, const int* __restrict__ plab,
    float* __restrict__ acc) {
  __shared__ float sh[256];
  __shared__ float shdiag;
  const int i = blockIdx.x;
  const int tid = threadIdx.x;
  if (tid == 0) shdiag = 0.f;
  __syncthreads();

  const float ms = mean[i];
  const float ns = norm[i];
  const int   li = lab[i];

  float d1 = 0.f, dp = 0.f, possum = 0.f, poscnt = 0.f;
  for (int j = tid; j < B_ROWS; j += 256) {
    // L_c term: corr(s_i, t_j)
    float cov = D1[i * B_ROWS + j] - (float)C_COLS * ms * mean[B_ROWS + j];
    float S = fminf(1.f, fmaxf(-1.f, cov / (ns * norm[B_ROWS + j] + EPSF)));
    float St = S / TAU;
    d1 += expf(St);
    if (j == i) shdiag = St;
    // L_p term: corr(s_i, p_j) / T
    float covp = D2[i * B_ROWS + j] - (float)C_COLS * ms * mean[2 * B_ROWS + j];
    float Sp = fminf(1.f, fmaxf(-1.f, covp / (ns * norm[2 * B_ROWS + j] + EPSF))) / TAU;
    dp += expf(Sp);
    if (plab[j] == li) { possum += Sp; poscnt += 1.f; }
  }
  d1     = block_sum(d1, sh);
  dp     = block_sum(dp, sh);
  possum = block_sum(possum, sh);
  poscnt = block_sum(poscnt, sh);
  if (tid == 0) {
    float Lc = logf(d1 + EPSF) - shdiag;                          // -log(num/(denom+eps))
    float Lp = (poscnt > 0.f) ? (logf(dp + EPSF) - possum / poscnt) : 0.f;
    atomicAdd(&acc[0], Lc);
    atomicAdd(&acc[1], Lp);
  }
}

// ---------------- Stage D: L_intra (per-class pearson of batch-softmax columns) ----------------
__global__ __launch_bounds__(256) void intra_kernel(
    const float* __restrict__ s, const float* __restrict__ t, float* __restrict__ acc) {
  __shared__ float sh[256];
  const int col = blockIdx.x * 256 + threadIdx.x;
  float corr = 0.f;
  if (col < C_COLS) {
    float mxs = -3.0e38f, mxt = -3.0e38f;
    for (int b = 0; b < B_ROWS; ++b) {
      mxs = fmaxf(mxs, s[(size_t)b * C_COLS + col]);
      mxt = fmaxf(mxt, t[(size_t)b * C_COLS + col]);
    }
    mxs *= 0.25f; mxt *= 0.25f;        // max of x/T
    float Zs = 0.f, Zt = 0.f, ss = 0.f, tt = 0.f, st = 0.f;
    for (int b = 0; b < B_ROWS; ++b) {
      float es = expf(s[(size_t)b * C_COLS + col] * 0.25f - mxs);
      float et = expf(t[(size_t)b * C_COLS + col] * 0.25f - mxt);
      Zs += es; Zt += et;
      ss += es * es; tt += et * et; st += es * et;
    }
    // softmax columns have mean exactly 1/B
    const float invB = 1.f / (float)B_ROWS;
    float cov = st / (Zs * Zt) - invB;
    float ns2 = fmaxf(ss / (Zs * Zs) - invB, 0.f);
    float nt2 = fmaxf(tt / (Zt * Zt) - invB, 0.f);
    corr = fminf(1.f, fmaxf(-1.f, cov / (sqrtf(ns2) * sqrtf(nt2) + EPSF)));
  }
  float bs = block_sum(corr, sh);
  if (threadIdx.x == 0) atomicAdd(&acc[2], bs);
}

// ---------------- Stage E: L_kd = KL(teacher || student), log-target form ----------------
__global__ __launch_bounds__(256) void kd_kernel(
    const float* __restrict__ s, const float* __restrict__ t, float* __restrict__ acc) {
  __shared__ float sh[256];
  const int i = blockIdx.x;
  const float* Sr = s + (size_t)i * C_COLS;
  const float* Tr = t + (size_t)i * C_COLS;

  float ma = -3.0e38f, mb = -3.0e38f;
  for (int c = threadIdx.x; c < C_COLS; c += 256) {
    ma = fmaxf(ma, Tr[c]);
    mb = fmaxf(mb, Sr[c]);
  }
  ma = block_max(ma, sh) * 0.25f;      // max of t/T
  mb = block_max(mb, sh) * 0.25f;      // max of s/T

  float za = 0.f, zb = 0.f, w = 0.f;
  for (int c = threadIdx.x; c < C_COLS; c += 256) {
    float a = Tr[c] * 0.25f;
    float b = Sr[c] * 0.25f;
    float ea = expf(a - ma);
    za += ea;
    zb += expf(b - mb);
    w  += ea * (a - b);
  }
  za = block_sum(za, sh);
  zb = block_sum(zb, sh);
  w  = block_sum(w, sh);
  if (threadIdx.x == 0) {
    // sum_c p_t * (log p_t - log p_s) = W/Za - LSE(a) + LSE(b)
    float kd = w / za - (ma + logf(za)) + (mb + logf(zb));
    atomicAdd(&acc[3], kd);
  }
}

// ---------------- Finalize ----------------
__global__ void finalize_kernel(const float* __restrict__ acc, float* __restrict__ out) {
  float Lc     = acc[0] / (float)B_ROWS;
  float Lp     = acc[1] / (float)B_ROWS;
  float Lintra = 1.f - (TAU * TAU) * (acc[2] / (float)C_COLS);
  float Lkd    = acc[3] * (TAU * TAU) / ((float)B_ROWS * (float)C_COLS);
  out[0] = Lkd + (Lc + Lp) + Lintra;
}

// ---------------- Host launcher ----------------
extern "C" void kernel_launch(void* const* d_in, const int* in_sizes, int n_in,
                              void* d_out, int out_size, void* d_ws, size_t ws_size,
                              hipStream_t stream) {
  const float* s    = (const float*)d_in[0];   // outputs_kd          [B, C]
  const float* t    = (const float*)d_in[1];   // teacher_outputs     [B, C]
  const int*   lab  = (const int*)d_in[2];     // original_labels     [B]
  const float* p    = (const float*)d_in[3];   // prev_teacher_logits [B, C]
  const int*   plab = (const int*)d_in[4];     // prev_teacher_labels [B]
  float* out = (float*)d_out;

  float* ws   = (float*)d_ws;
  float* mean = ws;                       // 3*1024 floats (s, t, p)
  float* norm = ws + 3 * B_ROWS;          // 3*1024 floats
  float* acc  = ws + 6 * B_ROWS;          // 8 floats
  float* D1   = ws + 8192;                // 1024*1024 floats : s . t^T
  float* D2   = D1 + (size_t)B_ROWS * B_ROWS;   // 1024*1024 floats : s . p^T

  hipMemsetAsync(acc, 0, 8 * sizeof(float), stream);

  row_stats_kernel<<<dim3(B_ROWS, 3), 256, 0, stream>>>(s, t, p, mean, norm);

  gemm_nt_wmma<<<dim3(B_ROWS / TN, B_ROWS / TM), 256, 0, stream>>>(s, t, D1, C_COLS);
  gemm_nt_wmma<<<dim3(B_ROWS / TN, B_ROWS / TM), 256, 0, stream>>>(s, p, D2, C_COLS);

  contrast_kernel<<<B_ROWS, 256, 0, stream>>>(D1, D2, mean, norm, lab, plab, acc);
  intra_kernel<<<(C_COLS + 255) / 256, 256, 0, stream>>>(s, t, acc);
  kd_kernel<<<B_ROWS, 256, 0, stream>>>(s, t, acc);
  finalize_kernel<<<1, 1, 0, stream>>>(acc, out);
}